// RDF2Sys_7550552507060
// MI455X (gfx1250) — compile-verified
//
#include <hip/hip_runtime.h>
#include <math.h>

#define BOXF      25.0f
#define HALF_BOXF 12.5f
#define NBINS     400
#define NP1       512
#define NP2       512
#define NPAIRS    (NP1 * NP2)                 // 262144
#define CHUNKS    64
#define PAIRS_PER_CHUNK (NPAIRS / CHUNKS)     // 4096
#define WAVES_PER_BLOCK 8
#define PAIRS_PER_WAVE  (PAIRS_PER_CHUNK / WAVES_PER_BLOCK)  // 512
#define BIN_TILES (NBINS / 16)                // 25

typedef float v2f __attribute__((ext_vector_type(2)));
typedef float v8f __attribute__((ext_vector_type(8)));

// One block: 8 wave32s. Each wave handles 512 pairs x one 16-bin tile.
// Stage 1: masked min-image distances -> LDS.
// Stage 2: V_WMMA_F32_16X16X4_F32 accumulation of the Gaussian soft-histogram.
__global__ __launch_bounds__(256)
void rdf_hist_kernel(const float* __restrict__ pos1,
                     const float* __restrict__ pos2,
                     float* __restrict__ partial)
{
    __shared__ float sd[WAVES_PER_BLOCK][PAIRS_PER_WAVE];  // 16 KB
    __shared__ float red[WAVES_PER_BLOCK * 16];

    const int tid   = (int)threadIdx.x;
    const int lane  = tid & 31;
    const int wave  = tid >> 5;
    const int tile  = (int)blockIdx.x;   // 0..24 -> bins [tile*16, tile*16+16)
    const int chunk = (int)blockIdx.y;   // 0..63 -> pairs [chunk*4096, ...)

    const int wave_base = chunk * PAIRS_PER_CHUNK + wave * PAIRS_PER_WAVE;

    // ---- Stage 1: distances with mask folded in (masked -> +inf -> gauss 0)
    #pragma unroll 4
    for (int it = 0; it < PAIRS_PER_WAVE / 32; ++it) {
        int p = wave_base + it * 32 + lane;
        int i = p & (NP1 - 1);   // pos1 index (fastest)
        int j = p >> 9;          // pos2 index
        float dx = pos1[3 * i + 0] - pos2[3 * j + 0];
        float dy = pos1[3 * i + 1] - pos2[3 * j + 1];
        float dz = pos1[3 * i + 2] - pos2[3 * j + 2];
        // python-style remainder(diff + 12.5, 25) - 12.5
        dx += HALF_BOXF; dx -= floorf(dx * (1.0f / BOXF)) * BOXF; dx -= HALF_BOXF;
        dy += HALF_BOXF; dy -= floorf(dy * (1.0f / BOXF)) * BOXF; dy -= HALF_BOXF;
        dz += HALF_BOXF; dz -= floorf(dz * (1.0f / BOXF)) * BOXF; dz -= HALF_BOXF;
        float d = sqrtf(dx * dx + dy * dy + dz * dz);
        sd[wave][it * 32 + lane] = (d > 1.0f) ? d : __builtin_inff();
    }
    __syncthreads();

    // ---- Stage 2: WMMA-accumulated Gaussian histogram
    const float step  = HALF_BOXF / (float)(NBINS - 1);   // linspace step (width)
    const float coeff = -0.5f / (step * step);
    const float o     = (float)(tile * 16 + (lane & 15)) * step;  // this lane's bin center (M row)
    const int   ksel  = 2 * (lane >> 4);                  // A-matrix K base for this half-wave

    const v2f bones = {1.0f, 1.0f};   // B = ones: D[m,n] = sum_k A[m,k] (layout-independent)
    v8f c = {0.f, 0.f, 0.f, 0.f, 0.f, 0.f, 0.f, 0.f};

    for (int s = 0; s < PAIRS_PER_WAVE / 4; ++s) {
        const float* sp = &sd[wave][s * 4 + ksel];
        float t0 = sp[0] - o;
        float t1 = sp[1] - o;
        v2f a;
        a.x = __expf(coeff * t0 * t0);   // A[M=lane%16, K=ksel]
        a.y = __expf(coeff * t1 * t1);   // A[M=lane%16, K=ksel+1]
        c = __builtin_amdgcn_wmma_f32_16x16x4_f32(
                /*neg_a=*/false, a, /*neg_b=*/false, bones,
                /*c_mod=*/(short)0, c, /*reuse_a=*/false, /*reuse_b=*/false);
    }

    // D column N=0 carries the per-bin sums: lane 0 -> M=0..7, lane 16 -> M=8..15
    if (lane == 0) {
        #pragma unroll
        for (int r = 0; r < 8; ++r) red[wave * 16 + r] = c[r];
    }
    if (lane == 16) {
        #pragma unroll
        for (int r = 0; r < 8; ++r) red[wave * 16 + 8 + r] = c[r];
    }
    __syncthreads();

    // fixed-order cross-wave reduce -> one partial vector per (chunk, bin)
    if (tid < 16) {
        float s = 0.0f;
        #pragma unroll
        for (int w = 0; w < WAVES_PER_BLOCK; ++w) s += red[w * 16 + tid];
        partial[chunk * NBINS + tile * 16 + tid] = s;
    }
}

// Single block: fixed-order chunk reduction, normalize, emit count|bins|rdf.
__global__ __launch_bounds__(512)
void rdf_finalize_kernel(const float* __restrict__ partial,
                         float* __restrict__ out)
{
    __shared__ float cr[NBINS];
    __shared__ float stotal;
    const int b = (int)threadIdx.x;

    if (b < NBINS) {
        float s = 0.0f;
        for (int ch = 0; ch < CHUNKS; ++ch) s += partial[ch * NBINS + b];
        cr[b] = s;
    }
    __syncthreads();
    if (b == 0) {
        float t = 0.0f;
        for (int i = 0; i < NBINS; ++i) t += cr[i];
        stotal = t;
    }
    __syncthreads();

    if (b <= NBINS) {  // bins[401]
        out[NBINS + b] = HALF_BOXF * (float)b / (float)NBINS;
    }
    if (b < NBINS) {
        float cnt = cr[b] / stotal;
        out[b] = cnt;  // count[400]
        float lo = HALF_BOXF * (float)b / (float)NBINS;
        float hi = HALF_BOXF * (float)(b + 1) / (float)NBINS;
        // rdf = count / (vol_bins / V); (4pi/3) cancels
        float volr = (hi * hi * hi - lo * lo * lo) /
                     (HALF_BOXF * HALF_BOXF * HALF_BOXF);
        out[2 * NBINS + 1 + b] = cnt / volr;  // rdf[400]
    }
}

extern "C" void kernel_launch(void* const* d_in, const int* in_sizes, int n_in,
                              void* d_out, int out_size, void* d_ws, size_t ws_size,
                              hipStream_t stream)
{
    (void)in_sizes; (void)n_in; (void)out_size; (void)ws_size;
    const float* pos1 = (const float*)d_in[0];
    const float* pos2 = (const float*)d_in[1];
    float* out = (float*)d_out;
    float* partial = (float*)d_ws;   // CHUNKS*NBINS floats = 100 KB, fully overwritten

    dim3 grid(BIN_TILES, CHUNKS);
    rdf_hist_kernel<<<grid, dim3(256), 0, stream>>>(pos1, pos2, partial);
    rdf_finalize_kernel<<<1, dim3(512), 0, stream>>>(partial, out);
}